// MambaSSD_26903675142797
// MI455X (gfx1250) — compile-verified
//
#include <hip/hip_runtime.h>

#define BATCH   128
#define NHEADS  64
#define HEADDIM 64
#define D_STATE 128

// clang ext vectors (needed for WMMA builtin operand types + nontemporal builtins)
typedef __attribute__((ext_vector_type(2))) float v2f;
typedef __attribute__((ext_vector_type(4))) float v4f;
typedef __attribute__((ext_vector_type(8))) float v8f;

// LDS row pitch for the 64x32 partial-sum matrix. 36 floats:
//  - stage-1 writes lds_part[p*36 + lane]: fixed p, lane 0..31 -> consecutive banks, no conflict
//  - stage-2 64-bit reads at (m*36 + 4c + 2h): 36*l mod 64 tiles the 64 banks in 4-bank blocks,
//    lanes l and l+16 land in the same block at disjoint offsets -> conflict-free
#define LDS_PITCH 36

__global__ __launch_bounds__(256)
void mamba_ssd_step(const float* __restrict__ x_heads,   // (B,H,P)
                    const float* __restrict__ B_in,      // (B,N)
                    const float* __restrict__ C_in,      // (B,N)
                    const float* __restrict__ dt_raw,    // (B,H)
                    const float* __restrict__ state,     // (B,H,P,N)
                    const float* __restrict__ A_log,     // (H)
                    const float* __restrict__ Dvec,      // (H)
                    const float* __restrict__ dt_bias,   // (H)
                    float* __restrict__ y_out,           // (B,H,P)
                    float* __restrict__ ns_out)          // (B,H,P,N)
{
    __shared__ float lds_part[HEADDIM * LDS_PITCH];   // 9216 B
    __shared__ float lds_x[HEADDIM];

    const int tid  = threadIdx.x;
    const int lane = tid & 31;
    const int wave = tid >> 5;
    const int bh   = blockIdx.x;          // b*NHEADS + h
    const int b    = bh >> 6;
    const int h    = bh & (NHEADS - 1);

    // Per-(b,h) scalars; uniform across the block (broadcast scalar loads).
    const float z   = dt_raw[bh] + dt_bias[h];
    const float dtv = (z > 20.0f) ? z : log1pf(expf(z));     // softplus
    const float dA  = expf(dtv * (-expf(A_log[h])));
    const float Dh  = Dvec[h];

    // Lane l owns state columns 4l..4l+3 for every row it touches.
    const v4f Bv = *(const v4f*)(B_in + b * D_STATE + 4 * lane);
    const v4f Cv = *(const v4f*)(C_in + b * D_STATE + 4 * lane);

    if (tid < HEADDIM) lds_x[tid] = x_heads[(size_t)bh * HEADDIM + tid];
    __syncthreads();

    const size_t sbase = (size_t)bh * (HEADDIM * D_STATE);
    const float* __restrict__ sin  = state  + sbase;
    float*       __restrict__ sout = ns_out + sbase;

    // ---- Stage 1: stream 64x128 state tile (wave w -> rows w*8..w*8+7) ----
    v4f s[8];
#pragma unroll
    for (int r = 0; r < 8; ++r) {
        const int p = wave * 8 + r;
        s[r] = __builtin_nontemporal_load((const v4f*)(sin + p * D_STATE + 4 * lane));
    }
#pragma unroll
    for (int r = 0; r < 8; ++r) {
        const int p   = wave * 8 + r;
        const float dtx = dtv * lds_x[p];
        v4f ns;
        ns.x = fmaf(s[r].x, dA, dtx * Bv.x);
        ns.y = fmaf(s[r].y, dA, dtx * Bv.y);
        ns.z = fmaf(s[r].z, dA, dtx * Bv.z);
        ns.w = fmaf(s[r].w, dA, dtx * Bv.w);
        __builtin_nontemporal_store(ns, (v4f*)(sout + p * D_STATE + 4 * lane));
        // partial dot with C over this lane's 4 columns
        lds_part[p * LDS_PITCH + lane] =
            fmaf(ns.x, Cv.x, fmaf(ns.y, Cv.y, fmaf(ns.z, Cv.z, ns.w * Cv.w)));
    }
    __syncthreads();

    // ---- Stage 2: y[p] = sum_lane partials  via V_WMMA_F32_16X16X4_F32 ----
    // Y(64x1) = PART(64x32) x ones(32x1); waves 0..3 each reduce one 16-row tile.
    // A (16x4 f32) ISA layout: lanes 0-15 hold M=lane, K={0,1} in v0,v1;
    //                          lanes 16-31 hold M=lane-16, K={2,3}.
    if (wave < 4) {
        const int m  = wave * 16 + (lane & 15);
        const int kb = (lane >> 4) * 2;
        const v2f ones = {1.0f, 1.0f};   // B = all ones -> every D column equals y
        v8f acc = {};
#pragma unroll
        for (int c = 0; c < 8; ++c) {    // K = 32 partials, 4 per WMMA
            v2f a = *(const v2f*)(&lds_part[m * LDS_PITCH + 4 * c + kb]);
            acc = __builtin_amdgcn_wmma_f32_16x16x4_f32(
                      false, a, false, ones, (short)0, acc, false, false);
        }
        // D layout: VGPR r, lanes 0-15 -> M=r, lanes 16-31 -> M=8+r (all N equal).
        if ((lane & 15) == 0) {
            const int p0 = wave * 16 + (lane >> 4) * 8;
            float* __restrict__ yrow = y_out + (size_t)bh * HEADDIM;
#pragma unroll
            for (int r = 0; r < 8; ++r)
                yrow[p0 + r] = fmaf(Dh, lds_x[p0 + r], acc[r]);
        }
    }
}

extern "C" void kernel_launch(void* const* d_in, const int* in_sizes, int n_in,
                              void* d_out, int out_size, void* d_ws, size_t ws_size,
                              hipStream_t stream) {
    const float* x_heads = (const float*)d_in[0];
    const float* B_in    = (const float*)d_in[1];
    const float* C_in    = (const float*)d_in[2];
    const float* dt_raw  = (const float*)d_in[3];
    const float* state   = (const float*)d_in[4];
    const float* A_log   = (const float*)d_in[5];
    const float* Dvec    = (const float*)d_in[6];
    const float* dt_bias = (const float*)d_in[7];

    float* y_out  = (float*)d_out;                                  // (B,H,P)
    float* ns_out = y_out + (size_t)BATCH * NHEADS * HEADDIM;       // (B,H,P,N)

    dim3 grid(BATCH * NHEADS);   // 8192 blocks, one per (b,h)
    dim3 block(256);             // 8 waves (wave32)
    mamba_ssd_step<<<grid, block, 0, stream>>>(
        x_heads, B_in, C_in, dt_raw, state, A_log, Dvec, dt_bias, y_out, ns_out);
}